// VectorQuantizerEMA_3496103379403
// MI455X (gfx1250) — compile-verified
//
#include <hip/hip_runtime.h>
#include <hip/hip_bf16.h>

typedef __attribute__((ext_vector_type(2))) float v2f;
typedef __attribute__((ext_vector_type(8))) float v8f;

#define NUM_CODES 1024
#define CODE_DIM  256
#define N_ROWS    32768
#define TEMP_INV  10.0f
#define DECAY     0.99f
#define OMD       0.01f
#define EPS_V     1e-5f

// output offsets (floats), reference tuple return order
#define OFF_ZQ    0
#define OFF_CODES 8388608
#define OFF_LOSS  8421376
#define OFF_PERP  8421377
#define OFF_ENT   8421378
#define OFF_SOFT  8421379
#define OFF_NEMB  41975811
#define OFF_NCS   42237955
#define OFF_NEA   42238979

// ---------------- per-code squared norms -> ws_e2 ----------------
__global__ void k_e2(const float* __restrict__ embed, float* __restrict__ e2) {
  int wave = threadIdx.x >> 5, lane = threadIdx.x & 31;
  int row  = blockIdx.x * 8 + wave;
  const float* p = embed + row * CODE_DIM;
  float s = 0.f;
  for (int i = lane; i < CODE_DIM; i += 32) { float v = p[i]; s += v * v; }
  #pragma unroll
  for (int off = 16; off; off >>= 1) s += __shfl_xor(s, off, 32);
  if (lane == 0) e2[row] = s;
}

// ---------------- main: WMMA GEMM + argmin + softmax + z_q ----------------
// 128 threads = 4 waves. Block owns 16 z-rows x all 1024 codes.
// Wave w owns column band [w*256, w*256+256) as 16 register-resident 16x16 tiles.
__global__ __launch_bounds__(128) void k_main(const float* __restrict__ z,
                                              const float* __restrict__ embed,
                                              const float* __restrict__ e2g,
                                              float* __restrict__ out,
                                              int* __restrict__ codes_ws,
                                              float* __restrict__ part_ws) {
  __shared__ float zt[16 * 260];     // padded stride: lane-strided ds reads hit distinct banks
  __shared__ float e2s[1024];
  __shared__ float wmin[4][16];
  __shared__ int   widx[4][16];
  __shared__ float wsum[4][16];
  __shared__ float dminS[16];
  __shared__ int   codeS[16];
  __shared__ float invZ[16];
  __shared__ float wpart[4];

  const int tid  = threadIdx.x;
  const int wave = tid >> 5, lane = tid & 31;
  const int nl   = lane & 15, hi = lane >> 4;
  const int rowbase = blockIdx.x * 16;

  // stage 16x256 z tile into LDS (coalesced float4)
  const float* zrowp = z + rowbase * CODE_DIM;
  #pragma unroll
  for (int i = 0; i < 8; ++i) {
    int off = (i * 128 + tid) * 4;
    int r = off >> 8, c = off & 255;
    float4 v = *(const float4*)(zrowp + off);
    *(float4*)&zt[r * 260 + c] = v;
  }
  #pragma unroll
  for (int i = 0; i < 8; ++i) e2s[i * 128 + tid] = e2g[i * 128 + tid];
  __syncthreads();

  // ---- GEMM: acc[j] = (z_tile) x (embed_band_j)^T via V_WMMA_F32_16X16X4_F32 ----
  v8f acc[16];
  #pragma unroll
  for (int j = 0; j < 16; ++j) acc[j] = v8f{0.f,0.f,0.f,0.f,0.f,0.f,0.f,0.f};

  const int colband = wave * 256;
  // ISA fp32 A 16x4 layout: lane holds (M = lane&15, K = 2*hi + v) -> contiguous pair in d
  const float* abase = &zt[nl * 260 + 2 * hi];
  // B 4x16 layout (mirrored): lane holds (K = 2*hi + v, N = lane&15); B[k,n] = embed[col_n][d+k]
  const float* bbase = embed + (colband + nl) * CODE_DIM + 2 * hi;

  for (int dc = 0; dc < 64; ++dc) {
    const int d = dc * 4;
    v2f a = *(const v2f*)(abase + d);           // ds_load_b64
    #pragma unroll
    for (int j = 0; j < 16; ++j) {
      v2f b = *(const v2f*)(bbase + j * 16 * CODE_DIM + d);   // global_load_b64 (L2-hot)
      acc[j] = __builtin_amdgcn_wmma_f32_16x16x4_f32(
          false, a, false, b, (short)0, acc[j], false, false);
    }
  }

  // ---- Phase A: dd = e2 - 2*acc ; per-row argmin (z^2 shift is argmin/softmax invariant) ----
  float minv[8]; int mini[8];
  #pragma unroll
  for (int r = 0; r < 8; ++r) { minv[r] = 3.4e38f; mini[r] = 0; }
  #pragma unroll
  for (int j = 0; j < 16; ++j) {
    int col = colband + j * 16 + nl;
    float e2v = e2s[col];
    #pragma unroll
    for (int r = 0; r < 8; ++r) {
      float ddv = e2v - 2.0f * acc[j][r];       // C/D layout: row = r + 8*hi, col as above
      acc[j][r] = ddv;
      if (ddv < minv[r]) { minv[r] = ddv; mini[r] = col; }
    }
  }
  #pragma unroll
  for (int r = 0; r < 8; ++r) {
    #pragma unroll
    for (int off = 1; off < 16; off <<= 1) {
      float ov = __shfl_xor(minv[r], off, 32);
      int   oi = __shfl_xor(mini[r], off, 32);
      if (ov < minv[r] || (ov == minv[r] && oi < mini[r])) { minv[r] = ov; mini[r] = oi; }
    }
    if (nl == 0) { wmin[wave][r + 8 * hi] = minv[r]; widx[wave][r + 8 * hi] = mini[r]; }
  }
  __syncthreads();
  if (tid < 16) {
    float bm = wmin[0][tid]; int bi = widx[0][tid];
    #pragma unroll
    for (int w = 1; w < 4; ++w) {
      float v = wmin[w][tid]; int ii = widx[w][tid];
      if (v < bm || (v == bm && ii < bi)) { bm = v; bi = ii; }
    }
    dminS[tid] = bm; codeS[tid] = bi;
  }
  __syncthreads();

  // ---- Phase B: stable softmax over K (exp values kept in acc regs) ----
  float rs[8], dml[8];
  #pragma unroll
  for (int r = 0; r < 8; ++r) { rs[r] = 0.f; dml[r] = dminS[r + 8 * hi]; }
  #pragma unroll
  for (int j = 0; j < 16; ++j) {
    #pragma unroll
    for (int r = 0; r < 8; ++r) {
      float e = __expf((dml[r] - acc[j][r]) * TEMP_INV);   // <= 1, no overflow
      acc[j][r] = e;
      rs[r] += e;
    }
  }
  #pragma unroll
  for (int r = 0; r < 8; ++r) {
    #pragma unroll
    for (int off = 1; off < 16; off <<= 1) rs[r] += __shfl_xor(rs[r], off, 32);
    if (nl == 0) wsum[wave][r + 8 * hi] = rs[r];
  }
  __syncthreads();
  if (tid < 16) invZ[tid] = 1.0f / (wsum[0][tid] + wsum[1][tid] + wsum[2][tid] + wsum[3][tid]);
  __syncthreads();

  // ---- Phase C: write soft_assign (dominant 134 MB stream) ----
  float izl[8];
  #pragma unroll
  for (int r = 0; r < 8; ++r) izl[r] = invZ[r + 8 * hi];
  float* softp = out + OFF_SOFT;
  #pragma unroll
  for (int j = 0; j < 16; ++j) {
    int col = colband + j * 16 + nl;
    #pragma unroll
    for (int r = 0; r < 8; ++r) {
      int row = rowbase + r + 8 * hi;
      softp[row * 1024 + col] = acc[j][r] * izl[r];
    }
  }

  // ---- Phase D: codes, z_q_st (== embed[code]), commitment partial ----
  if (tid < 16) {
    out[OFF_CODES + rowbase + tid] = (float)codeS[tid];
    codes_ws[rowbase + tid] = codeS[tid];
  }
  float cpart = 0.f;
  #pragma unroll
  for (int q = 0; q < 4; ++q) {
    int m = wave * 4 + q;
    int code = codeS[m];
    const float* ep = embed + code * CODE_DIM;
    float* zqp = out + OFF_ZQ + (rowbase + m) * CODE_DIM;
    #pragma unroll
    for (int i = 0; i < 8; ++i) {
      int d2 = lane + i * 32;
      float ev = ep[d2];
      float zv = zt[m * 260 + d2];
      zqp[d2] = ev;
      float df = zv - ev;
      cpart += df * df;
    }
  }
  #pragma unroll
  for (int off = 16; off; off >>= 1) cpart += __shfl_xor(cpart, off, 32);
  if (lane == 0) wpart[wave] = cpart;
  __syncthreads();
  if (tid == 0) part_ws[blockIdx.x] = wpart[0] + wpart[1] + wpart[2] + wpart[3];
}

// ---------------- code histogram (integer LDS atomics: deterministic) ----------------
__global__ void k_counts(const int* __restrict__ codes, float* __restrict__ counts) {
  __shared__ int cnt[NUM_CODES];
  int t = threadIdx.x;                    // 1024 threads
  cnt[t] = 0;
  __syncthreads();
  for (int i = 0; i < 32; ++i) atomicAdd(&cnt[codes[i * 1024 + t]], 1);
  __syncthreads();
  counts[t] = (float)cnt[t];
}

// ---------------- deterministic segment sums + EMA buffers ----------------
// 32 blocks x 256 threads; block owns 32 codes, thread owns column d, n scanned ascending.
__global__ void k_segsum(const float* __restrict__ z, const int* __restrict__ codes,
                         const float* __restrict__ counts,
                         const float* __restrict__ cluster_size,
                         const float* __restrict__ embed_avg,
                         float* __restrict__ out) {
  __shared__ float accs[32][CODE_DIM];
  int d = threadIdx.x;
  int k0 = blockIdx.x * 32;
  #pragma unroll
  for (int r = 0; r < 32; ++r) accs[r][d] = 0.f;
  __syncthreads();
  for (int n = 0; n < N_ROWS; ++n) {
    int c = codes[n];                      // uniform -> scalar-cached
    unsigned rr = (unsigned)(c - k0);
    if (rr < 32u) accs[rr][d] += z[n * CODE_DIM + d];
  }
  __syncthreads();
  for (int r = 0; r < 32; ++r) {
    int k = k0 + r;
    out[OFF_NEA + k * CODE_DIM + d] =
        DECAY * embed_avg[k * CODE_DIM + d] + OMD * accs[r][d];
  }
  if (d < 32) {
    int k = k0 + d;
    out[OFF_NCS + k] = DECAY * cluster_size[k] + OMD * counts[k];
  }
}

// ---------------- scalar reductions: n_total, entropy, perplexity, loss ----------------
__global__ void k_scalars(const float* __restrict__ counts,
                          const float* __restrict__ part,
                          float* __restrict__ out, float* __restrict__ ws_n) {
  __shared__ float sred[256];
  int t = threadIdx.x;
  float s = 0.f;
  for (int i = t; i < NUM_CODES; i += 256) s += out[OFF_NCS + i];
  sred[t] = s; __syncthreads();
  for (int o = 128; o; o >>= 1) { if (t < o) sred[t] += sred[t + o]; __syncthreads(); }
  if (t == 0) ws_n[0] = sred[0];
  __syncthreads();

  float e = 0.f;
  for (int i = t; i < NUM_CODES; i += 256) {
    float p = counts[i] / 32768.0f;
    e -= p * __logf(p + 1e-10f);
  }
  sred[t] = e; __syncthreads();
  for (int o = 128; o; o >>= 1) { if (t < o) sred[t] += sred[t + o]; __syncthreads(); }
  float ent = sred[0];
  __syncthreads();

  float c = 0.f;
  for (int i = t; i < 2048; i += 256) c += part[i];
  sred[t] = c; __syncthreads();
  for (int o = 128; o; o >>= 1) { if (t < o) sred[t] += sred[t + o]; __syncthreads(); }
  if (t == 0) {
    out[OFF_LOSS] = sred[0] / (32768.0f * 256.0f);
    out[OFF_ENT]  = ent;
    out[OFF_PERP] = __expf(ent);
  }
}

// ---------------- new_embed = new_embed_avg / cs_norm ----------------
__global__ void k_newembed(const float* __restrict__ ws_n, float* __restrict__ out) {
  int k = blockIdx.x, d = threadIdx.x;
  float n = ws_n[0];
  float ncs = out[OFF_NCS + k];
  float csn = (ncs + EPS_V) / (n + 1024.0f * EPS_V) * n;
  out[OFF_NEMB + k * CODE_DIM + d] = out[OFF_NEA + k * CODE_DIM + d] / csn;
}

extern "C" void kernel_launch(void* const* d_in, const int* in_sizes, int n_in,
                              void* d_out, int out_size, void* d_ws, size_t ws_size,
                              hipStream_t stream) {
  const float* z            = (const float*)d_in[0];
  const float* embed        = (const float*)d_in[1];
  const float* cluster_size = (const float*)d_in[2];
  const float* embed_avg    = (const float*)d_in[3];
  float* out = (float*)d_out;

  float* wsf        = (float*)d_ws;
  float* ws_e2      = wsf;                    // 1024
  int*   ws_codes   = (int*)(wsf + 1024);     // 32768
  float* ws_counts  = wsf + 1024 + 32768;     // 1024
  float* ws_partial = ws_counts + 1024;       // 2048
  float* ws_n       = ws_partial + 2048;      // 1

  k_e2      <<<128, 256, 0, stream>>>(embed, ws_e2);
  k_main    <<<2048, 128, 0, stream>>>(z, embed, ws_e2, out, ws_codes, ws_partial);
  k_counts  <<<1, 1024, 0, stream>>>(ws_codes, ws_counts);
  k_segsum  <<<32, 256, 0, stream>>>(z, ws_codes, ws_counts, cluster_size, embed_avg, out);
  k_scalars <<<1, 256, 0, stream>>>(ws_counts, ws_partial, out, ws_n);
  k_newembed<<<1024, 256, 0, stream>>>(ws_n, out);
}